// MMDLoss_85822036508767
// MI455X (gfx1250) — compile-verified
//
#include <hip/hip_runtime.h>
#include <hip/hip_bf16.h>

// ---------------------------------------------------------------------------
// MMD loss, fused:  source/target 4096x256 f32 -> scalar f32
//   pass 1: row norms sq[i]
//   pass 2: bandwidth via  sum(l2) = 2*(ns*sum(sq) - ||colsum||^2)   (O(ns*D))
//   pass 3: fused f32-WMMA Gram GEMM + multi-bandwidth Gaussian epilogue,
//           UPPER-TRIANGULAR tiles only (symmetry: off-diagonal weight 2)
//   pass 4: deterministic tree reduction
// ---------------------------------------------------------------------------

typedef __attribute__((ext_vector_type(2))) float v2f;
typedef __attribute__((ext_vector_type(8))) float v8f;

#define DDIM   256
#define NROWS  4096
#define NS     8192
#define TILE   32
#define TPD    (NS / TILE)                 // 256 tiles per dimension
#define NTILES (TPD * (TPD + 1) / 2)       // 32896 upper-triangular tiles
#define NBLK_MAIN (NTILES / 8)             // 4112 blocks (8 waves -> 8 tiles)

// workspace layout (floats)
#define WS_SQ   0                 // [0 .. NS-1]       row squared norms
#define WS_C0   NS                // [NS]              exp coefficient
#define WS_PART (NS + 1)          // [NS+1 ..]         per-block partials

// ---------------------------------------------------------------------------
// Kernel 1: sq[r] = ||row r||^2.  One wave (32 lanes) per row.
// ---------------------------------------------------------------------------
__global__ void __launch_bounds__(256) sq_kernel(const float* __restrict__ src,
                                                 const float* __restrict__ tgt,
                                                 float* __restrict__ ws) {
    const int wave = threadIdx.x >> 5;
    const int lane = threadIdx.x & 31;
    const int row  = blockIdx.x * 8 + wave;              // 0..8191
    const float* base = (row < NROWS) ? (src + row * DDIM)
                                      : (tgt + (row - NROWS) * DDIM);
    const float4* p = (const float4*)(base + lane * 8);  // 8 floats per lane
    float4 a = p[0], b = p[1];
    float s = a.x*a.x + a.y*a.y + a.z*a.z + a.w*a.w
            + b.x*b.x + b.y*b.y + b.z*b.z + b.w*b.w;
#pragma unroll
    for (int m = 16; m >= 1; m >>= 1) s += __shfl_xor(s, m, 32);
    if (lane == 0) ws[WS_SQ + row] = s;
}

// ---------------------------------------------------------------------------
// Kernel 2: bandwidth coefficient.  Single block, 256 threads (one per column).
//   sum_l2 = 2*(NS*sum(sq) - ||colsum||^2)
//   bw     = sum_l2/(NS^2-NS) / 4        (KERNEL_MUL^(KERNEL_NUM//2) = 4)
//   c0     = log2(e)/bw      so that  exp(-l2/(bw*2^t)) = exp2(-l2*c0*2^-t)
// ---------------------------------------------------------------------------
__global__ void __launch_bounds__(256) bw_kernel(const float* __restrict__ src,
                                                 const float* __restrict__ tgt,
                                                 float* __restrict__ ws) {
    __shared__ float redA[256];
    __shared__ float redB[256];
    const int c = threadIdx.x;
    float cs = 0.f;
    for (int r = 0; r < NROWS; ++r) cs += src[r * DDIM + c];
    for (int r = 0; r < NROWS; ++r) cs += tgt[r * DDIM + c];
    float sqs = 0.f;
#pragma unroll
    for (int k = 0; k < NS / DDIM; ++k) sqs += ws[WS_SQ + c + k * DDIM];
    redA[c] = cs * cs;   // partial ||colsum||^2
    redB[c] = sqs;       // partial sum(sq)
    __syncthreads();
    for (int off = 128; off >= 1; off >>= 1) {
        if (c < off) { redA[c] += redA[c + off]; redB[c] += redB[c + off]; }
        __syncthreads();
    }
    if (c == 0) {
        double s2    = (double)redA[0];
        double sqsum = (double)redB[0];
        double suml2 = 2.0 * ((double)NS * sqsum - s2);
        double bw    = suml2 / ((double)NS * (double)NS - (double)NS);
        bw *= 0.25;  // / KERNEL_MUL^(KERNEL_NUM//2)
        ws[WS_C0] = (float)(1.4426950408889634 / bw);
    }
}

// 5-bandwidth Gaussian kernel sum; bandwidths double -> halve the exponent arg
__device__ __forceinline__ float ksum5(float l2, float c0) {
    float a = l2 * c0;
    float s = exp2f(-a);
    s += exp2f(-a * 0.5f);
    s += exp2f(-a * 0.25f);
    s += exp2f(-a * 0.125f);
    s += exp2f(-a * 0.0625f);
    return s;
}

// start index of tile-row ti in the packed upper-triangular ordering
__device__ __forceinline__ int tri_start(int ti) {
    return ti * TPD - (ti * (ti - 1)) / 2;
}

// ---------------------------------------------------------------------------
// Kernel 3: fused Gram GEMM (f32 WMMA 16x16x4) + epilogue.
// Each wave computes one 32x32 output tile = 2x2 WMMA accumulators.
// Only upper-triangular tiles (tj >= ti); off-diagonal tiles weighted 2x.
// A-matrix lane layout (f32 16x4): lanes 0-15 -> K=0,1 ; lanes 16-31 -> K=2,3.
// C/D layout: row = v + 8*(lane>=16), col = lane&15.
// ---------------------------------------------------------------------------
__global__ void __launch_bounds__(256, 2) mmd_main(const float* __restrict__ src,
                                                   const float* __restrict__ tgt,
                                                   float* __restrict__ ws) {
    __shared__ float red[256];
    const int lane = threadIdx.x & 31;
    const int wave = threadIdx.x >> 5;
    const int t    = blockIdx.x * 8 + wave;  // 0..NTILES-1

    // decode packed upper-triangular index -> (ti, tj), tj >= ti
    int ti = (int)((513.0f - sqrtf(513.0f * 513.0f - 8.0f * (float)t)) * 0.5f);
    if (ti > TPD - 1) ti = TPD - 1;
    if (ti < 0) ti = 0;
    while (ti + 1 <= TPD - 1 && tri_start(ti + 1) <= t) ++ti;  // exact fixup
    while (tri_start(ti) > t) --ti;
    const int tj = ti + (t - tri_start(ti));

    const int i0 = ti * TILE;
    const int j0 = tj * TILE;

    // 4096 is a multiple of 32, so each tile's rows come from a single input
    const float* pa = (i0 < NROWS) ? (src + i0 * DDIM) : (tgt + (i0 - NROWS) * DDIM);
    const float* pb = (j0 < NROWS) ? (src + j0 * DDIM) : (tgt + (j0 - NROWS) * DDIM);
    const float sgn = ((i0 < NROWS) ? 1.f : -1.f) * ((j0 < NROWS) ? 1.f : -1.f);
    const float wgt = (ti == tj) ? 1.f : 2.f;   // symmetry weight

    const int r  = lane & 15;         // matrix row within 16 for A/B fragments
    const int kb = (lane >> 4) * 2;   // K offset within a 4-wide chunk

    const float* a0p = pa + r * DDIM + kb;
    const float* a1p = a0p + 16 * DDIM;
    const float* b0p = pb + r * DDIM + kb;
    const float* b1p = b0p + 16 * DDIM;

    v8f acc00 = {}, acc01 = {}, acc10 = {}, acc11 = {};

#pragma unroll 4
    for (int kc = 0; kc < DDIM; kc += 4) {
        v2f a0 = *(const v2f*)(a0p + kc);
        v2f a1 = *(const v2f*)(a1p + kc);
        v2f b0 = *(const v2f*)(b0p + kc);
        v2f b1 = *(const v2f*)(b1p + kc);
        acc00 = __builtin_amdgcn_wmma_f32_16x16x4_f32(false, a0, false, b0,
                                                      (short)0, acc00, false, false);
        acc01 = __builtin_amdgcn_wmma_f32_16x16x4_f32(false, a0, false, b1,
                                                      (short)0, acc01, false, false);
        acc10 = __builtin_amdgcn_wmma_f32_16x16x4_f32(false, a1, false, b0,
                                                      (short)0, acc10, false, false);
        acc11 = __builtin_amdgcn_wmma_f32_16x16x4_f32(false, a1, false, b1,
                                                      (short)0, acc11, false, false);
    }

    // ---- epilogue: l2 -> 5-bandwidth kernel sum, signed accumulate ----
    const float  c0 = ws[WS_C0];
    const float* sq = ws + WS_SQ;
    const int rowBase = (lane >> 4) * 8;     // C/D row offset within 16
    const float sqj0 = sq[j0 + (lane & 15)];
    const float sqj1 = sq[j0 + 16 + (lane & 15)];

    float part = 0.f;
#pragma unroll
    for (int v = 0; v < 8; ++v) {
        const float sqi0 = sq[i0 + rowBase + v];
        const float sqi1 = sq[i0 + 16 + rowBase + v];
        part += ksum5(sqi0 + sqj0 - 2.f * acc00[v], c0);
        part += ksum5(sqi0 + sqj1 - 2.f * acc01[v], c0);
        part += ksum5(sqi1 + sqj0 - 2.f * acc10[v], c0);
        part += ksum5(sqi1 + sqj1 - 2.f * acc11[v], c0);
    }
    part *= sgn * wgt;

    red[threadIdx.x] = part;
    __syncthreads();
    for (int off = 128; off >= 1; off >>= 1) {
        if (threadIdx.x < off) red[threadIdx.x] += red[threadIdx.x + off];
        __syncthreads();
    }
    if (threadIdx.x == 0) ws[WS_PART + blockIdx.x] = red[0];
}

// ---------------------------------------------------------------------------
// Kernel 4: deterministic final reduction -> d_out[0]
// ---------------------------------------------------------------------------
__global__ void __launch_bounds__(256) finalize_kernel(const float* __restrict__ ws,
                                                       float* __restrict__ out) {
    __shared__ float red[256];
    float s = 0.f;
    for (int k = threadIdx.x; k < NBLK_MAIN; k += 256) s += ws[WS_PART + k];
    red[threadIdx.x] = s;
    __syncthreads();
    for (int off = 128; off >= 1; off >>= 1) {
        if (threadIdx.x < off) red[threadIdx.x] += red[threadIdx.x + off];
        __syncthreads();
    }
    if (threadIdx.x == 0)
        out[0] = red[0] * (1.0f / ((float)NROWS * (float)NROWS));
}

extern "C" void kernel_launch(void* const* d_in, const int* in_sizes, int n_in,
                              void* d_out, int out_size, void* d_ws, size_t ws_size,
                              hipStream_t stream) {
    const float* src = (const float*)d_in[0];
    const float* tgt = (const float*)d_in[1];
    float* ws  = (float*)d_ws;
    float* out = (float*)d_out;

    sq_kernel      <<<NS / 8,     256, 0, stream>>>(src, tgt, ws);
    bw_kernel      <<<1,          256, 0, stream>>>(src, tgt, ws);
    mmd_main       <<<NBLK_MAIN,  256, 0, stream>>>(src, tgt, ws);
    finalize_kernel<<<1,          256, 0, stream>>>(ws, out);
}